// GNNModel_54193897341485
// MI455X (gfx1250) — compile-verified
//
#include <hip/hip_runtime.h>
#include <hip/hip_bf16.h>
#include <math.h>

typedef float v2f __attribute__((ext_vector_type(2)));
typedef float v8f __attribute__((ext_vector_type(8)));

#define F_IN 128
#define HDIM 64
#define BN_EPS 1e-5f

// ---------------------------------------------------------------------------
// Fold BatchNorm (+ pre-BN bias) into per-feature scale/shift:
//   bn(y + b) = y*s + (b - mean)*s + beta,  s = g * rsqrt(var + eps)
// ---------------------------------------------------------------------------
__global__ void fold_bn_kernel(const float* __restrict__ g, const float* __restrict__ beta,
                               const float* __restrict__ mean, const float* __restrict__ var,
                               const float* __restrict__ b,
                               float* __restrict__ scale, float* __restrict__ shift, int n) {
    int i = blockIdx.x * blockDim.x + threadIdx.x;
    if (i < n) {
        float s = g[i] * rsqrtf(var[i] + BN_EPS);
        scale[i] = s;
        shift[i] = beta[i] + (b[i] - mean[i]) * s;
    }
}

// ---------------------------------------------------------------------------
// Degree / D^-1/2 (self-loops included: deg starts at 1)
// ---------------------------------------------------------------------------
__global__ void deg_init_kernel(float* __restrict__ deg, int N) {
    int i = blockIdx.x * blockDim.x + threadIdx.x;
    if (i < N) deg[i] = 1.0f;
}
__global__ void deg_accum_kernel(const int* __restrict__ col, float* __restrict__ deg, int E) {
    int i = blockIdx.x * blockDim.x + threadIdx.x;
    if (i < E) atomicAdd(&deg[col[i]], 1.0f);
}
__global__ void deg_to_dinv_kernel(float* __restrict__ deg, int N) {
    int i = blockIdx.x * blockDim.x + threadIdx.x;
    if (i < N) deg[i] = rsqrtf(deg[i]);   // deg >= 1 always
}

// ---------------------------------------------------------------------------
// Encoder GEMM: h = relu(bn(x @ enc_W + b))   [N,128] x [128,64]
// Block = 128 threads (4 waves). Block computes 16 rows; wave w does cols
// [16w, 16w+16). fp32 WMMA 16x16x4, K-loop of 32 steps. A tile staged in LDS.
// ---------------------------------------------------------------------------
__global__ __launch_bounds__(128)
void encoder_gemm_kernel(const float* __restrict__ x, const float* __restrict__ W,
                         const float* __restrict__ scale, const float* __restrict__ shift,
                         float* __restrict__ h, int N) {
    const int lane = threadIdx.x & 31;
    const int wave = threadIdx.x >> 5;
    const int row0 = blockIdx.x * 16;
    const int col0 = wave * 16;
    const int m  = lane & 15;
    const int kh = (lane >> 4) << 1;           // 0 or 2

    __shared__ float xa[16][F_IN + 2];         // +2 pad: avoid 16-way bank conflict
    for (int i = threadIdx.x; i < 16 * F_IN; i += 128) {
        int rr = i >> 7, kk = i & (F_IN - 1);
        int gr = row0 + rr;
        xa[rr][kk] = (gr < N) ? x[(size_t)gr * F_IN + kk] : 0.0f;
    }
    __syncthreads();

    v8f acc = {};
    for (int k = 0; k < F_IN; k += 4) {
        v2f a, b;
        a.x = xa[m][k + kh];
        a.y = xa[m][k + kh + 1];
        b.x = W[(k + kh) * HDIM + col0 + m];
        b.y = W[(k + kh + 1) * HDIM + col0 + m];
        acc = __builtin_amdgcn_wmma_f32_16x16x4_f32(false, a, false, b,
                                                    (short)0, acc, false, false);
    }
    const int n = col0 + m;
    const float sc = scale[n], sh = shift[n];
    for (int r = 0; r < 8; ++r) {
        int mm = r + ((lane >> 4) << 3);
        int gr = row0 + mm;
        if (gr < N) h[(size_t)gr * HDIM + n] = fmaxf(acc[r] * sc + sh, 0.0f);
    }
}

// ---------------------------------------------------------------------------
// GCN dense part: hw = h @ W   [N,64] x [64,64]
// ---------------------------------------------------------------------------
__global__ __launch_bounds__(128)
void gcn_gemm_kernel(const float* __restrict__ hin, const float* __restrict__ W,
                     float* __restrict__ hw, int N) {
    const int lane = threadIdx.x & 31;
    const int wave = threadIdx.x >> 5;
    const int row0 = blockIdx.x * 16;
    const int col0 = wave * 16;
    const int m  = lane & 15;
    const int kh = (lane >> 4) << 1;

    __shared__ float ha[16][HDIM + 2];
    for (int i = threadIdx.x; i < 16 * HDIM; i += 128) {
        int rr = i >> 6, kk = i & (HDIM - 1);
        int gr = row0 + rr;
        ha[rr][kk] = (gr < N) ? hin[(size_t)gr * HDIM + kk] : 0.0f;
    }
    __syncthreads();

    v8f acc = {};
    for (int k = 0; k < HDIM; k += 4) {
        v2f a, b;
        a.x = ha[m][k + kh];
        a.y = ha[m][k + kh + 1];
        b.x = W[(k + kh) * HDIM + col0 + m];
        b.y = W[(k + kh + 1) * HDIM + col0 + m];
        acc = __builtin_amdgcn_wmma_f32_16x16x4_f32(false, a, false, b,
                                                    (short)0, acc, false, false);
    }
    for (int r = 0; r < 8; ++r) {
        int mm = r + ((lane >> 4) << 3);
        int gr = row0 + mm;
        if (gr < N) hw[(size_t)gr * HDIM + col0 + m] = acc[r];
    }
}

// ---------------------------------------------------------------------------
// Scatter: out = segment_sum(hw[row]*norm, col) + hw*dinv^2 (self loop) + bias
// ---------------------------------------------------------------------------
__global__ void scatter_init_kernel(const float* __restrict__ hw, const float* __restrict__ dinv,
                                    const float* __restrict__ bias, float* __restrict__ out, int N) {
    size_t i = (size_t)blockIdx.x * blockDim.x + threadIdx.x;
    if (i < (size_t)N * HDIM) {
        int node = (int)(i >> 6), f = (int)(i & (HDIM - 1));
        float d = dinv[node];
        out[i] = hw[i] * d * d + bias[f];
    }
}

__global__ void scatter_edges_kernel(const float* __restrict__ hw, const int* __restrict__ row,
                                     const int* __restrict__ col, const float* __restrict__ dinv,
                                     float* __restrict__ out, int E) {
    size_t t = (size_t)blockIdx.x * blockDim.x + threadIdx.x;
    int e = (int)(t >> 5);                 // one wave per edge, 2 feats/lane
    if (e >= E) return;
    int f = ((int)t & 31) * 2;
    int r = row[e], c = col[e];            // scalar per wave -> broadcast loads
    float nrm = dinv[r] * dinv[c];
    const float2 v = *(const float2*)(hw + (size_t)r * HDIM + f);
    atomicAdd(&out[(size_t)c * HDIM + f],     v.x * nrm);
    atomicAdd(&out[(size_t)c * HDIM + f + 1], v.y * nrm);
}

__global__ void relu_kernel(float* __restrict__ p, size_t n) {
    size_t i = (size_t)blockIdx.x * blockDim.x + threadIdx.x;
    if (i < n) p[i] = fmaxf(p[i], 0.0f);
}

// ---------------------------------------------------------------------------
// Edge predictor: pred[e] = sigmoid( relu(bn(cat(h[row],h[col]) @ W1 + b1)) @ w2 + b2 )
// Block = 128 threads (4 waves), 16 edges per block. Gathered 16x128 A tile in
// LDS, 32 fp32-WMMA steps per wave (16 cols each), ds_add_f32 dot reduction.
// ---------------------------------------------------------------------------
__global__ __launch_bounds__(128)
void edge_pred_kernel(const float* __restrict__ h, const int* __restrict__ row,
                      const int* __restrict__ col, const float* __restrict__ W1,
                      const float* __restrict__ scale, const float* __restrict__ shift,
                      const float* __restrict__ w2, const float* __restrict__ b2,
                      float* __restrict__ pred, int E) {
    const int lane = threadIdx.x & 31;
    const int wave = threadIdx.x >> 5;
    const int e0   = blockIdx.x * 16;
    const int col0 = wave * 16;
    const int m  = lane & 15;
    const int kh = (lane >> 4) << 1;

    __shared__ float ef[16][2 * HDIM + 2];
    __shared__ float acc16[16];

    for (int i = threadIdx.x; i < 16 * 2 * HDIM; i += 128) {
        int rr = i >> 7, kk = i & (2 * HDIM - 1);
        int e = e0 + rr;
        float v = 0.0f;
        if (e < E) {
            int node = (kk < HDIM) ? row[e] : col[e];
            v = h[(size_t)node * HDIM + (kk & (HDIM - 1))];
        }
        ef[rr][kk] = v;
    }
    if (threadIdx.x < 16) acc16[threadIdx.x] = 0.0f;
    __syncthreads();

    v8f acc = {};
    for (int k = 0; k < 2 * HDIM; k += 4) {
        v2f a, b;
        a.x = ef[m][k + kh];
        a.y = ef[m][k + kh + 1];
        b.x = W1[(k + kh) * HDIM + col0 + m];
        b.y = W1[(k + kh + 1) * HDIM + col0 + m];
        acc = __builtin_amdgcn_wmma_f32_16x16x4_f32(false, a, false, b,
                                                    (short)0, acc, false, false);
    }
    const int n = col0 + m;
    const float sc = scale[n], sh = shift[n], wv = w2[n];
    for (int r = 0; r < 8; ++r) {
        int mm = r + ((lane >> 4) << 3);
        float v = fmaxf(acc[r] * sc + sh, 0.0f);   // bn-fold + relu
        atomicAdd(&acc16[mm], v * wv);             // ds_add_f32
    }
    __syncthreads();
    if (threadIdx.x < 16) {
        int e = e0 + threadIdx.x;
        if (e < E) {
            float z = acc16[threadIdx.x] + b2[0];
            pred[e] = 1.0f / (1.0f + expf(-z));
        }
    }
}

// ---------------------------------------------------------------------------
extern "C" void kernel_launch(void* const* d_in, const int* in_sizes, int n_in,
                              void* d_out, int out_size, void* d_ws, size_t ws_size,
                              hipStream_t stream) {
    const float* x        = (const float*)d_in[0];
    const int*   eidx     = (const int*)  d_in[1];
    const float* enc_W    = (const float*)d_in[2];
    const float* enc_b    = (const float*)d_in[3];
    const float* enc_g    = (const float*)d_in[4];
    const float* enc_beta = (const float*)d_in[5];
    const float* enc_mean = (const float*)d_in[6];
    const float* enc_var  = (const float*)d_in[7];
    const float* conv1_W  = (const float*)d_in[8];
    const float* conv1_b  = (const float*)d_in[9];
    const float* conv2_W  = (const float*)d_in[10];
    const float* conv2_b  = (const float*)d_in[11];
    const float* ep1_W    = (const float*)d_in[12];
    const float* ep1_b    = (const float*)d_in[13];
    const float* ep_g     = (const float*)d_in[14];
    const float* ep_beta  = (const float*)d_in[15];
    const float* ep_mean  = (const float*)d_in[16];
    const float* ep_var   = (const float*)d_in[17];
    const float* ep2_W    = (const float*)d_in[18];
    const float* ep2_b    = (const float*)d_in[19];

    const int N = in_sizes[0] / F_IN;
    const int E = in_sizes[1] / 2;
    const int* row = eidx;
    const int* col = eidx + E;

    float* out_nodes = (float*)d_out;                    // [N,64]
    float* out_pred  = out_nodes + (size_t)N * HDIM;     // [E]

    // workspace: dinv[N] | hw[N*64] | enc scale/shift | ep scale/shift
    float* ws        = (float*)d_ws;
    float* dinv      = ws;
    float* hw        = ws + (((size_t)N + 127) & ~(size_t)127);
    float* enc_scale = hw + (size_t)N * HDIM;
    float* enc_shift = enc_scale + HDIM;
    float* ep_scale  = enc_shift + HDIM;
    float* ep_shift  = ep_scale + HDIM;

    const int TPB = 256;
    const size_t NH = (size_t)N * HDIM;

    // BN folds
    fold_bn_kernel<<<1, HDIM, 0, stream>>>(enc_g, enc_beta, enc_mean, enc_var, enc_b,
                                           enc_scale, enc_shift, HDIM);
    fold_bn_kernel<<<1, HDIM, 0, stream>>>(ep_g, ep_beta, ep_mean, ep_var, ep1_b,
                                           ep_scale, ep_shift, HDIM);

    // degree -> dinv
    deg_init_kernel<<<(N + TPB - 1) / TPB, TPB, 0, stream>>>(dinv, N);
    deg_accum_kernel<<<(E + TPB - 1) / TPB, TPB, 0, stream>>>(col, dinv, E);
    deg_to_dinv_kernel<<<(N + TPB - 1) / TPB, TPB, 0, stream>>>(dinv, N);

    const int gemm_blocks = (N + 15) / 16;

    // node encoder -> h0 lives in d_out node region
    encoder_gemm_kernel<<<gemm_blocks, 128, 0, stream>>>(x, enc_W, enc_scale, enc_shift,
                                                         out_nodes, N);

    // conv1
    gcn_gemm_kernel<<<gemm_blocks, 128, 0, stream>>>(out_nodes, conv1_W, hw, N);
    scatter_init_kernel<<<(int)((NH + TPB - 1) / TPB), TPB, 0, stream>>>(hw, dinv, conv1_b,
                                                                         out_nodes, N);
    scatter_edges_kernel<<<(int)(((size_t)E * 32 + TPB - 1) / TPB), TPB, 0, stream>>>(
        hw, row, col, dinv, out_nodes, E);
    relu_kernel<<<(int)((NH + TPB - 1) / TPB), TPB, 0, stream>>>(out_nodes, NH);

    // conv2
    gcn_gemm_kernel<<<gemm_blocks, 128, 0, stream>>>(out_nodes, conv2_W, hw, N);
    scatter_init_kernel<<<(int)((NH + TPB - 1) / TPB), TPB, 0, stream>>>(hw, dinv, conv2_b,
                                                                         out_nodes, N);
    scatter_edges_kernel<<<(int)(((size_t)E * 32 + TPB - 1) / TPB), TPB, 0, stream>>>(
        hw, row, col, dinv, out_nodes, E);
    relu_kernel<<<(int)((NH + TPB - 1) / TPB), TPB, 0, stream>>>(out_nodes, NH);

    // edge predictor
    edge_pred_kernel<<<(E + 15) / 16, 128, 0, stream>>>(out_nodes, row, col, ep1_W,
                                                        ep_scale, ep_shift, ep2_W, ep2_b,
                                                        out_pred, E);
}